// junction_48447231098948
// MI455X (gfx1250) — compile-verified
//
#include <hip/hip_runtime.h>
#include <hip/hip_bf16.h>

// Problem constants
#define BB 8
#define NN 256
#define CC 512
#define HW 64
#define DIM 1024
#define M_TOT (BB * NN)          // 2048 rows
#define ROWLEN (CC * HW)         // 32768 floats per (b,n)

typedef float v2f __attribute__((ext_vector_type(2)));
typedef float v8f __attribute__((ext_vector_type(8)));

// ---------------------------------------------------------------------------
// Kernel 1: one block per (b,n). Single streaming pass over x (256 MB total):
//   z[m, c*2+o] = sum_h x[m,c,h] * Ws[c,o,h]   (unscaled)
//   s[m]        = sum_{c,h} x[m,c,h]
// Each half-wave (16 lanes) handles one channel per iteration; lanes load
// float4 -> 512 contiguous bytes per wave access (coalesced).
// ---------------------------------------------------------------------------
__global__ void __launch_bounds__(256)
k_reduce_contract(const float* __restrict__ x, const float* __restrict__ Ws,
                  float* __restrict__ z, float* __restrict__ s)
{
    const int m    = blockIdx.x;          // 0..2047
    const int tid  = threadIdx.x;         // 0..255
    const int wave = tid >> 5;            // 0..7 (wave32)
    const int lane = tid & 31;
    const int half = lane >> 4;           // 0/1: which channel of the pair
    const int lr   = lane & 15;           // lane within half

    const float* xb = x + (size_t)m * ROWLEN;
    float acc = 0.0f;
    __shared__ float wsums[8];

    #pragma unroll 4
    for (int it = 0; it < 32; ++it) {
        const int c = wave * 64 + it * 2 + half;      // channel handled by this half
        const float4 xv = *(const float4*)(xb + c * HW + lr * 4);
        const float4 w0 = *(const float4*)(Ws + (size_t)(c * 2 + 0) * HW + lr * 4);
        const float4 w1 = *(const float4*)(Ws + (size_t)(c * 2 + 1) * HW + lr * 4);

        float p0 = xv.x * w0.x + xv.y * w0.y + xv.z * w0.z + xv.w * w0.w;
        float p1 = xv.x * w1.x + xv.y * w1.y + xv.z * w1.z + xv.w * w1.w;
        float ps = xv.x + xv.y + xv.z + xv.w;

        #pragma unroll
        for (int off = 1; off < 16; off <<= 1) {      // butterfly within half
            p0 += __shfl_xor(p0, off, 32);
            p1 += __shfl_xor(p1, off, 32);
            ps += __shfl_xor(ps, off, 32);
        }
        acc += ps;                                    // every lane holds its half's channel sum
        if (lr == 0) {
            z[(size_t)m * DIM + c * 2 + 0] = p0;
            z[(size_t)m * DIM + c * 2 + 1] = p1;
        }
    }
    acc += __shfl_xor(acc, 16, 32);                   // combine both halves -> 64-channel sum
    if (lane == 0) wsums[wave] = acc;
    __syncthreads();
    if (tid == 0) {
        float t = 0.0f;
        #pragma unroll
        for (int w = 0; w < 8; ++w) t += wsums[w];
        s[m] = t;
    }
}

// ---------------------------------------------------------------------------
// Kernel 2: attention recurrence on the [B,N] means (2048 values, 1 block).
//   m0 = s / (C*HW);  3x: a = sigmoid(conv1d_same(m, w5) + b);  m *= a
//   scale = a0*a1*a2
// Thread t owns n=t for all 8 batches (k = batch index).
// ---------------------------------------------------------------------------
__global__ void __launch_bounds__(256)
k_attention(const float* __restrict__ s, const float* __restrict__ conv_w,
            const float* __restrict__ conv_b, float* __restrict__ scale)
{
    __shared__ float mbuf[M_TOT];
    const int tid = threadIdx.x;          // 0..255 == n
    float sc[BB];
    #pragma unroll
    for (int k = 0; k < BB; ++k) {
        const int idx = tid + k * NN;
        mbuf[idx] = s[idx] * (1.0f / (float)ROWLEN);
        sc[k] = 1.0f;
    }
    __syncthreads();

    for (int it = 0; it < 3; ++it) {
        const float w0 = conv_w[it * 5 + 0], w1 = conv_w[it * 5 + 1];
        const float w2 = conv_w[it * 5 + 2], w3 = conv_w[it * 5 + 3];
        const float w4 = conv_w[it * 5 + 4];
        const float cb = conv_b[it];
        const int n = tid;
        float newv[BB];
        #pragma unroll
        for (int k = 0; k < BB; ++k) {
            const float* row = mbuf + k * NN;
            float a = cb + row[n] * w2;               // cross-correlation, pad=2
            if (n >= 2)   a += row[n - 2] * w0;
            if (n >= 1)   a += row[n - 1] * w1;
            if (n <= 254) a += row[n + 1] * w3;
            if (n <= 253) a += row[n + 2] * w4;
            const float sg = 1.0f / (1.0f + expf(-a));
            sc[k] *= sg;
            newv[k] = sg * row[n];
        }
        __syncthreads();
        #pragma unroll
        for (int k = 0; k < BB; ++k) mbuf[tid + k * NN] = newv[k];
        __syncthreads();
    }
    #pragma unroll
    for (int k = 0; k < BB; ++k) scale[tid + k * NN] = sc[k];
}

// ---------------------------------------------------------------------------
// Kernel 3: fold bs into the fc bias:  bias2[d] = fc_b[d] + sum_k bs[k]*fc_w[d,k]
// ---------------------------------------------------------------------------
__global__ void __launch_bounds__(256)
k_bias2(const float* __restrict__ bs, const float* __restrict__ fc_w,
        const float* __restrict__ fc_b, float* __restrict__ bias2)
{
    const int d = blockIdx.x * 256 + threadIdx.x;
    if (d >= DIM) return;
    const float* wr = fc_w + (size_t)d * DIM;
    float acc = fc_b[d];
    for (int k = 0; k < DIM; ++k) acc += bs[k] * wr[k];
    bias2[d] = acc;
}

// ---------------------------------------------------------------------------
// Kernel 4: out[m,d] = sum_k (scale[m]*z[m,k]) * fc_w[d,k] + bias2[d]
// Exact fp32 GEMM via V_WMMA_F32_16X16X4_F32 (wave32, one wave per block).
// Each wave computes a 16x64 strip: 4 accumulators, K stepped by 4.
// A (16x4 fp32) frag: lane (half,lr) -> row M=lr, elems K = k0+2*half+{0,1}.
// B (4x16 fp32) frag: lane (half,lr) -> col N=lr, elems K = k0+2*half+{0,1}.
// C/D layout per ISA: VGPR v, lane -> M = v + 8*half, N = lr.
// ---------------------------------------------------------------------------
__global__ void __launch_bounds__(32)
k_gemm_wmma(const float* __restrict__ z, const float* __restrict__ scale,
            const float* __restrict__ fc_w, const float* __restrict__ bias2,
            float* __restrict__ out)
{
    const int lane = threadIdx.x;
    const int half = lane >> 4;
    const int lr   = lane & 15;
    const int tm   = blockIdx.x * 16;     // output row tile
    const int d0   = blockIdx.y * 64;     // output col strip

    const float  sm    = scale[tm + lr];
    const float* zrow  = z    + (size_t)(tm + lr) * DIM + 2 * half;
    const float* wrow0 = fc_w + (size_t)(d0 + lr) * DIM + 2 * half;

    v8f acc[4];
    #pragma unroll
    for (int t = 0; t < 4; ++t)
        #pragma unroll
        for (int v = 0; v < 8; ++v) acc[t][v] = 0.0f;

    for (int k0 = 0; k0 < DIM; k0 += 4) {
        v2f a;
        a[0] = zrow[k0]     * sm;
        a[1] = zrow[k0 + 1] * sm;
        #pragma unroll
        for (int t = 0; t < 4; ++t) {
            const float* wr = wrow0 + (size_t)t * 16 * DIM;
            v2f b;
            b[0] = wr[k0];
            b[1] = wr[k0 + 1];
            acc[t] = __builtin_amdgcn_wmma_f32_16x16x4_f32(
                false, a, false, b, (short)0, acc[t], false, false);
        }
    }

    #pragma unroll
    for (int t = 0; t < 4; ++t) {
        const float bz = bias2[d0 + t * 16 + lr];
        #pragma unroll
        for (int v = 0; v < 8; ++v) {
            const int row = tm + v + 8 * half;
            out[(size_t)row * DIM + d0 + t * 16 + lr] = acc[t][v] + bz;
        }
    }
}

// ---------------------------------------------------------------------------
extern "C" void kernel_launch(void* const* d_in, const int* in_sizes, int n_in,
                              void* d_out, int out_size, void* d_ws, size_t ws_size,
                              hipStream_t stream)
{
    (void)in_sizes; (void)n_in; (void)out_size; (void)ws_size;
    const float* x      = (const float*)d_in[0];   // [8,256,512,8,8]
    const float* conv_w = (const float*)d_in[1];   // [3,5]
    const float* conv_b = (const float*)d_in[2];   // [3]
    const float* Ws     = (const float*)d_in[3];   // [512,2,64]
    const float* bs     = (const float*)d_in[4];   // [512,2]
    const float* fc_w   = (const float*)d_in[5];   // [1024,1024]
    const float* fc_b   = (const float*)d_in[6];   // [1024]
    float* out = (float*)d_out;                    // [8,256,1024]

    float* ws    = (float*)d_ws;
    float* z     = ws;                             // 2048*1024 floats (8 MB)
    float* s     = z + (size_t)M_TOT * DIM;        // 2048
    float* scale = s + M_TOT;                      // 2048
    float* bias2 = scale + M_TOT;                  // 1024

    k_reduce_contract<<<M_TOT, 256, 0, stream>>>(x, Ws, z, s);
    k_attention<<<1, 256, 0, stream>>>(s, conv_w, conv_b, scale);
    k_bias2<<<(DIM + 255) / 256, 256, 0, stream>>>(bs, fc_w, fc_b, bias2);
    k_gemm_wmma<<<dim3(M_TOT / 16, DIM / 64), 32, 0, stream>>>(z, scale, fc_w, bias2, out);
}